// Voxelization_86260123173788
// MI455X (gfx1250) — compile-verified
//
#include <hip/hip_runtime.h>

#define RES   32
#define R3    (RES * RES * RES)
#define BATCH 8
#define CH    64
#define NPTS  100000

// NPTS = 1562 * 64 + 32 : full 64-element WMMA chunks + one 32-element tail
#define FULL_CHUNKS 1562
#define TAIL_BASE   (FULL_CHUNKS * 64)      // 99968
#define TAIL_WAVE   (FULL_CHUNKS % 8)       // wave that owns the tail chunk

typedef float v2f __attribute__((ext_vector_type(2)));
typedef float v8f __attribute__((ext_vector_type(8)));

// ---------------------------------------------------------------------------
// K0: zero a float4 region (grid-stride)
// ---------------------------------------------------------------------------
__global__ void k_zero_f4(float4* __restrict__ p, long n4) {
    long i = (long)blockIdx.x * blockDim.x + threadIdx.x;
    long stride = (long)gridDim.x * blockDim.x;
    float4 z = make_float4(0.f, 0.f, 0.f, 0.f);
    for (; i < n4; i += stride) p[i] = z;
}

// ---------------------------------------------------------------------------
// K1: per-(batch,dim) coordinate sum via V_WMMA_F32_16X16X4_F32.
// A = ones(16x4)  =>  D[m,n] = sum_k B[k,n] + C[m,n]; every row of D is the
// per-column sum, so summing one full row of D gives the exact f32 total of
// all 64 B elements, regardless of the A/B VGPR layout.  Each lane therefore
// feeds a CONTIGUOUS float2 (one global_load_b64, no predication) per chunk.
// Tail (32 elements) = exactly one wave of bm.x with bm.y = 0, handled by a
// uniform branch so EXEC is all-ones at every WMMA.
// grid = (3, BATCH), block = 256 (8 waves), wave w takes chunks w, w+8, ...
// ---------------------------------------------------------------------------
__global__ void k_mean_sum_wmma(const float* __restrict__ coords,
                                float* __restrict__ mean_sum /* [B][4] */) {
    const int dim  = blockIdx.x;        // 0..2
    const int b    = blockIdx.y;        // 0..7
    const int wave = threadIdx.x >> 5;  // 0..7
    const int lane = threadIdx.x & 31;

    const float* __restrict__ src = coords + ((size_t)b * 3 + dim) * NPTS;

    v2f a; a.x = 1.0f; a.y = 1.0f;      // all-ones A matrix
    v8f c = {};                          // f32 accumulator (8 VGPRs)

    // Full chunks: unconditional contiguous 8-byte load per lane.
    for (int ch = wave; ch < FULL_CHUNKS; ch += 8) {
        const v2f bm = *(const v2f*)(src + ch * 64 + 2 * lane);
        c = __builtin_amdgcn_wmma_f32_16x16x4_f32(
                false, a, false, bm, (short)0, c, false, false);
    }

    // Tail chunk: 32 valid elements == one full wave of bm.x, bm.y = 0.
    if (wave == TAIL_WAVE) {
        v2f bm;
        bm.x = src[TAIL_BASE + lane];
        bm.y = 0.0f;
        c = __builtin_amdgcn_wmma_f32_16x16x4_f32(
                false, a, false, bm, (short)0, c, false, false);
    }

    // VGPR0 lanes 0..15 hold one full row of D (N = 0..15).
    float v = (lane < 16) ? c[0] : 0.0f;
    for (int off = 16; off >= 1; off >>= 1)
        v += __shfl_xor(v, off, 32);
    if (lane == 0)
        atomicAdd(&mean_sum[b * 4 + dim], v);
}

// ---------------------------------------------------------------------------
// K2: per-batch max radius.  Max over r^2 then a single sqrt (monotone).
// Non-negative floats => uint bit pattern is order-preserving => atomicMax.
// grid = (128, BATCH), block = 256
// ---------------------------------------------------------------------------
__global__ void k_radius(const float* __restrict__ coords,
                         const float* __restrict__ mean_sum,
                         unsigned* __restrict__ radius_bits) {
    const int b = blockIdx.y;
    const float inv_n = 1.0f / (float)NPTS;
    const float mx = mean_sum[b * 4 + 0] * inv_n;
    const float my = mean_sum[b * 4 + 1] * inv_n;
    const float mz = mean_sum[b * 4 + 2] * inv_n;

    const float* __restrict__ cb = coords + (size_t)b * 3 * NPTS;
    float r2max = 0.0f;
    for (int n = blockIdx.x * blockDim.x + threadIdx.x; n < NPTS;
         n += gridDim.x * blockDim.x) {
        const float dx = cb[n] - mx;
        const float dy = cb[NPTS + n] - my;
        const float dz = cb[2 * NPTS + n] - mz;
        r2max = fmaxf(r2max, dx * dx + dy * dy + dz * dz);
    }

    __shared__ float smax[256];
    smax[threadIdx.x] = r2max;
    __syncthreads();
    for (int st = 128; st >= 1; st >>= 1) {
        if ((int)threadIdx.x < st)
            smax[threadIdx.x] = fmaxf(smax[threadIdx.x], smax[threadIdx.x + st]);
        __syncthreads();
    }
    if (threadIdx.x == 0)
        atomicMax(&radius_bits[b], __float_as_uint(sqrtf(smax[0])));
}

// ---------------------------------------------------------------------------
// K3: per-point normalize + scatter.  Writes norm_coords (output #2) and
// atomically accumulates feature sums into d_out[B][C][R3] and counts in ws.
// Accumulation set (64 MB sums + 1 MB counts) is L2-resident (192 MB L2),
// so the 51M f32 atomics never touch HBM.
// grid = (ceil(N/256), BATCH), block = 256
// ---------------------------------------------------------------------------
__global__ void k_scatter(const float* __restrict__ coords,
                          const float* __restrict__ features,
                          const float* __restrict__ mean_sum,
                          const unsigned* __restrict__ radius_bits,
                          float* __restrict__ out_sums,   // [B][C][R3]
                          float* __restrict__ out_norm,   // [B][3][N]
                          float* __restrict__ counts) {   // [B][R3]
    const int b = blockIdx.y;
    const int n = blockIdx.x * blockDim.x + threadIdx.x;
    if (n >= NPTS) return;

    const float inv_n = 1.0f / (float)NPTS;
    const float mx = mean_sum[b * 4 + 0] * inv_n;
    const float my = mean_sum[b * 4 + 1] * inv_n;
    const float mz = mean_sum[b * 4 + 2] * inv_n;
    const float radius = __uint_as_float(radius_bits[b]);
    const float scale  = 1.0f / (radius * 2.0f);   // EPS == 0

    const float* __restrict__ cb = coords + (size_t)b * 3 * NPTS;
    float ncx = ((cb[n]            - mx) * scale + 0.5f) * (float)RES;
    float ncy = ((cb[NPTS + n]     - my) * scale + 0.5f) * (float)RES;
    float ncz = ((cb[2 * NPTS + n] - mz) * scale + 0.5f) * (float)RES;
    ncx = fminf(fmaxf(ncx, 0.0f), (float)(RES - 1));
    ncy = fminf(fmaxf(ncy, 0.0f), (float)(RES - 1));
    ncz = fminf(fmaxf(ncz, 0.0f), (float)(RES - 1));

    float* __restrict__ nb = out_norm + (size_t)b * 3 * NPTS;
    nb[n]            = ncx;
    nb[NPTS + n]     = ncy;
    nb[2 * NPTS + n] = ncz;

    const int vx = (int)rintf(ncx);   // round-half-even, matches jnp.round
    const int vy = (int)rintf(ncy);
    const int vz = (int)rintf(ncz);
    const int idx = (vx * RES + vy) * RES + vz;

    atomicAdd(&counts[(size_t)b * R3 + idx], 1.0f);

    const float* __restrict__ fb = features + (size_t)b * CH * NPTS;
    float* __restrict__ ob = out_sums + (size_t)b * CH * R3;
#pragma unroll 4
    for (int c = 0; c < CH; ++c)
        atomicAdd(&ob[(size_t)c * R3 + idx], fb[(size_t)c * NPTS + n]);
}

// ---------------------------------------------------------------------------
// K4: out = sum / max(cnt, 1)   (in place over d_out sums region)
// ---------------------------------------------------------------------------
__global__ void k_normalize(float* __restrict__ out_sums,
                            const float* __restrict__ counts) {
    const size_t total = (size_t)BATCH * CH * R3;
    size_t i = (size_t)blockIdx.x * blockDim.x + threadIdx.x;
    if (i >= total) return;
    const size_t b = i / ((size_t)CH * R3);
    const size_t v = i % (size_t)R3;
    const float cnt = counts[b * R3 + v];
    out_sums[i] = out_sums[i] / fmaxf(cnt, 1.0f);
}

// ---------------------------------------------------------------------------
extern "C" void kernel_launch(void* const* d_in, const int* in_sizes, int n_in,
                              void* d_out, int out_size, void* d_ws, size_t ws_size,
                              hipStream_t stream) {
    const float* features = (const float*)d_in[0];  // [8,64,100000]
    const float* coords   = (const float*)d_in[1];  // [8,3,100000]

    float* out_sums = (float*)d_out;                        // [8,64,32768]
    float* out_norm = out_sums + (size_t)BATCH * CH * R3;   // [8,3,100000]

    // ws layout (floats): counts [B*R3] | mean_sum [B*4] | radius_bits [B]
    float*    counts      = (float*)d_ws;
    float*    mean_sum    = counts + (size_t)BATCH * R3;
    unsigned* radius_bits = (unsigned*)(mean_sum + BATCH * 4);

    // K0: zero accumulation regions (d_out/d_ws are poisoned by harness)
    {
        const long n4 = (long)BATCH * CH * R3 / 4;          // 4,194,304
        k_zero_f4<<<4096, 256, 0, stream>>>((float4*)out_sums, n4);
        const long w4 = ((long)BATCH * R3 + BATCH * 4 + BATCH) / 4;  // 65,546
        k_zero_f4<<<260, 256, 0, stream>>>((float4*)d_ws, w4);
    }

    // K1: per-(b,dim) coordinate sums on the WMMA pipe
    k_mean_sum_wmma<<<dim3(3, BATCH), 256, 0, stream>>>(coords, mean_sum);

    // K2: per-batch max radius
    k_radius<<<dim3(128, BATCH), 256, 0, stream>>>(coords, mean_sum, radius_bits);

    // K3: normalize coords + scatter-accumulate features/counts
    k_scatter<<<dim3((NPTS + 255) / 256, BATCH), 256, 0, stream>>>(
        coords, features, mean_sum, radius_bits, out_sums, out_norm, counts);

    // K4: divide by counts
    {
        const size_t total = (size_t)BATCH * CH * R3;       // 16,777,216
        k_normalize<<<(unsigned)((total + 255) / 256), 256, 0, stream>>>(
            out_sums, counts);
    }
}